// SWEGNN_19490561589351
// MI455X (gfx1250) — compile-verified
//
#include <hip/hip_runtime.h>

#define S_FEAT 16
#define D_FEAT 8
#define E_FEAT 4
#define EDGE_IN 52   // E_FEAT + 2*S_FEAT + 2*D_FEAT
#define HIDDEN 16    // 2*D_FEAT

typedef __attribute__((ext_vector_type(2))) float v2f;
typedef __attribute__((ext_vector_type(8))) float v8f;

// ---------------------------------------------------------------------------
// Fused kernel: edge MLP (fp32 WMMA 16x16x4) + per-edge L2 normalize +
// masked hydraulic-gradient scatter into acc, all in one pass so s_ij never
// touches HBM. One wave32 per 16-edge tile.
// ---------------------------------------------------------------------------
__global__ __launch_bounds__(256)
void swegnn_mlp_prop(const float* __restrict__ x_s,
                     const float* __restrict__ x_d,
                     const float* __restrict__ ea,
                     const float* __restrict__ W1,
                     const float* __restrict__ b1,
                     const float* __restrict__ W2,
                     const float* __restrict__ b2,
                     const int*   __restrict__ row,
                     const int*   __restrict__ col,
                     const float* __restrict__ out,   // current node state [N,8]
                     float*       __restrict__ acc,   // scatter target [N,8]
                     int nEdges)
{
    __shared__ float ldsE[8][16 * EDGE_IN];  // 16 x 52 feature tile per wave
    __shared__ float ldsH[8][16 * 17];       // 16 x 16 hidden tile (padded rows)
    __shared__ int   ldsRC[8][32];           // row[0..15], col[0..15]

    const int lane = threadIdx.x & 31;
    const int wv   = threadIdx.x >> 5;
    const int e0   = (blockIdx.x * 8 + wv) * 16;
    if (e0 >= nEdges) return;

    float* Et = ldsE[wv];
    float* Ht = ldsH[wv];
    int*   RC = ldsRC[wv];

    // stage row/col indices for this tile (clamped for safety on tails)
    if (lane < 16) {
        int ge = e0 + lane;
        if (ge >= nEdges) ge = nEdges - 1;
        RC[lane]      = row[ge];
        RC[lane + 16] = col[ge];
    }

    // cooperative gather of the 16 x 52 edge-feature tile into LDS
    for (int i = lane; i < 16 * EDGE_IN; i += 32) {
        int e  = i / EDGE_IN;
        int f  = i - e * EDGE_IN;
        int ge = e0 + e;
        if (ge >= nEdges) ge = nEdges - 1;
        int r = RC[e];
        int c = RC[e + 16];
        float v;
        if (f < 16)      v = x_s[r * S_FEAT + f];
        else if (f < 32) v = x_s[c * S_FEAT + (f - 16)];
        else if (f < 40) v = x_d[r * D_FEAT + (f - 32)];
        else if (f < 48) v = x_d[c * D_FEAT + (f - 40)];
        else             v = ea[(long long)ge * E_FEAT + (f - 48)];
        Et[e * EDGE_IN + f] = v;
    }
    // single-wave tile: per-wave in-order LDS makes this RAW-safe, no barrier

    const int n    = lane & 15;        // N (and A's M) index
    const int hi   = lane >> 4;        // which 16-lane half
    const int koff = hi * 2;           // f32 WMMA: half-wave K offset

    // ---- GEMM1: h[16x16] = Etile[16x52] x W1^T, K tiled by 4 ----
    v8f h = {};
#pragma unroll
    for (int kc = 0; kc < 13; ++kc) {
        int kb = kc * 4 + koff;
        v2f a, b;
        a.x = Et[n * EDGE_IN + kb];
        a.y = Et[n * EDGE_IN + kb + 1];
        b.x = W1[n * EDGE_IN + kb];        // B[k][n] = W1[n][k]
        b.y = W1[n * EDGE_IN + kb + 1];
        h = __builtin_amdgcn_wmma_f32_16x16x4_f32(false, a, false, b,
                                                  (short)0, h, false, false);
    }

    // bias + ReLU in C layout (N = lane&15, M = v + 8*hi), transpose via LDS
    float bias1 = b1[n];
#pragma unroll
    for (int v = 0; v < 8; ++v) {
        float hv = fmaxf(h[v] + bias1, 0.0f);
        Ht[(v + 8 * hi) * 17 + n] = hv;
    }

    // ---- GEMM2: s[16x16] = h[16x16] x W2^T(zero-padded to 16 cols) ----
    v8f s = {};
#pragma unroll
    for (int kc = 0; kc < 4; ++kc) {
        int kb = kc * 4 + koff;
        v2f a, b;
        a.x = Ht[n * 17 + kb];
        a.y = Ht[n * 17 + kb + 1];
        b.x = (n < 8) ? W2[n * HIDDEN + kb]     : 0.0f;
        b.y = (n < 8) ? W2[n * HIDDEN + kb + 1] : 0.0f;
        s = __builtin_amdgcn_wmma_f32_16x16x4_f32(false, a, false, b,
                                                  (short)0, s, false, false);
    }

    // bias2 + per-edge L2 norm (padded cols are exact zeros -> reduce the half)
    float bias2 = (n < 8) ? b2[n] : 0.0f;
    float sval[8];
#pragma unroll
    for (int v = 0; v < 8; ++v) {
        float x  = (n < 8) ? (s[v] + bias2) : 0.0f;
        float ss = x * x;
        ss += __shfl_xor(ss, 1, 32);
        ss += __shfl_xor(ss, 2, 32);
        ss += __shfl_xor(ss, 4, 32);
        ss += __shfl_xor(ss, 8, 32);
        float nrm = sqrtf(ss);
        sval[v] = (nrm > 0.0f) ? (x / nrm) : 0.0f;
    }

    // ---- fused propagation: masked (out[col]-out[row]) * s -> atomic acc ----
    // Lane n<8 owns feature n of edge m = v + 8*hi.
#pragma unroll
    for (int v = 0; v < 8; ++v) {
        int  m     = v + 8 * hi;
        int  ge    = e0 + m;
        bool valid = (ge < nEdges) && (n < 8);
        int  r = RC[m];
        int  c = RC[m + 16];
        float o_r = valid ? out[(long long)r * D_FEAT + n] : 0.0f;
        float o_c = valid ? out[(long long)c * D_FEAT + n] : 0.0f;
        // node-row sums over 8 features = reduction within the 8-lane group
        float sr = o_r, sc = o_c;
        sr += __shfl_xor(sr, 1, 32);  sc += __shfl_xor(sc, 1, 32);
        sr += __shfl_xor(sr, 2, 32);  sc += __shfl_xor(sc, 2, 32);
        sr += __shfl_xor(sr, 4, 32);  sc += __shfl_xor(sc, 4, 32);
        bool maskon = (sr != 0.0f) || (sc != 0.0f);
        if (valid && maskon)
            unsafeAtomicAdd(acc + (long long)c * D_FEAT + n, (o_c - o_r) * sval[v]);
    }
}

// ---------------------------------------------------------------------------
// out = x_d @ Wf0^T ; acc = 0
// ---------------------------------------------------------------------------
__global__ __launch_bounds__(256)
void swegnn_init(const float* __restrict__ x_d,
                 const float* __restrict__ Wf0,
                 float* __restrict__ out,
                 float* __restrict__ acc,
                 int nNodes)
{
    int i = blockIdx.x * blockDim.x + threadIdx.x;
    if (i >= nNodes) return;
    float xd[D_FEAT];
#pragma unroll
    for (int j = 0; j < D_FEAT; ++j) xd[j] = x_d[(long long)i * D_FEAT + j];
#pragma unroll
    for (int f = 0; f < D_FEAT; ++f) {
        float acc0 = 0.0f;
#pragma unroll
        for (int j = 0; j < D_FEAT; ++j) acc0 += xd[j] * Wf0[f * D_FEAT + j];
        out[(long long)i * D_FEAT + f] = acc0;
        acc[(long long)i * D_FEAT + f] = 0.0f;
    }
}

// ---------------------------------------------------------------------------
// out += acc @ Wf^T ; acc = 0 (ready for next step)
// ---------------------------------------------------------------------------
__global__ __launch_bounds__(256)
void swegnn_update(float* __restrict__ out,
                   float* __restrict__ acc,
                   const float* __restrict__ Wf,
                   int nNodes)
{
    int i = blockIdx.x * blockDim.x + threadIdx.x;
    if (i >= nNodes) return;
    float a[D_FEAT];
#pragma unroll
    for (int j = 0; j < D_FEAT; ++j) {
        a[j] = acc[(long long)i * D_FEAT + j];
        acc[(long long)i * D_FEAT + j] = 0.0f;
    }
#pragma unroll
    for (int f = 0; f < D_FEAT; ++f) {
        float y = 0.0f;
#pragma unroll
        for (int j = 0; j < D_FEAT; ++j) y += a[j] * Wf[f * D_FEAT + j];
        out[(long long)i * D_FEAT + f] += y;
    }
}

// ---------------------------------------------------------------------------
extern "C" void kernel_launch(void* const* d_in, const int* in_sizes, int n_in,
                              void* d_out, int out_size, void* d_ws, size_t ws_size,
                              hipStream_t stream) {
    const float* x_s = (const float*)d_in[0];
    const float* x_d = (const float*)d_in[1];
    const float* ea  = (const float*)d_in[2];
    const float* W1  = (const float*)d_in[3];
    const float* b1  = (const float*)d_in[4];
    const float* W2  = (const float*)d_in[5];
    const float* b2  = (const float*)d_in[6];
    const float* Wf0 = (const float*)d_in[7];
    const float* Wf1 = (const float*)d_in[8];
    const float* Wf2 = (const float*)d_in[9];
    const int*   ei  = (const int*)d_in[10];

    const int nEdges = in_sizes[2] / E_FEAT;   // edge_attr is [E, 4]
    const int nNodes = in_sizes[1] / D_FEAT;   // x_d is [N, 8]
    const int* rowp = ei;
    const int* colp = ei + nEdges;

    float* out = (float*)d_out;
    float* acc = (float*)d_ws;                 // [N, 8] scatter accumulator

    dim3 blk(256);
    dim3 nodeGrid((nNodes + 255) / 256);
    dim3 mlpGrid((nEdges + 127) / 128);        // 8 waves/block * 16 edges/wave

    swegnn_init<<<nodeGrid, blk, 0, stream>>>(x_d, Wf0, out, acc, nNodes);
    for (int k = 0; k < 2; ++k) {
        const float* Wf = (k == 0) ? Wf1 : Wf2;
        swegnn_mlp_prop<<<mlpGrid, blk, 0, stream>>>(x_s, x_d, ea, W1, b1, W2, b2,
                                                     rowp, colp, out, acc, nEdges);
        swegnn_update<<<nodeGrid, blk, 0, stream>>>(out, acc, Wf, nNodes);
    }
}